// GCN_66228395704919
// MI455X (gfx1250) — compile-verified
//
#include <hip/hip_runtime.h>

// ---------------- problem constants (match reference) ----------------
#define NNODES 50000
#define NEDGES 800000
#define FEATD  512
#define HIDD   256
#define CLSD   40
#define EPSBN  1e-5f

static constexpr int NCHUNK = (NNODES + 255) / 256;   // 196
static constexpr int MTILES = NNODES / 16;            // 3125 (exact)

typedef float v2f __attribute__((ext_vector_type(2)));
typedef float v8f __attribute__((ext_vector_type(8)));

__device__ __forceinline__ float4 ld4(const float* p) { return *(const float4*)p; }

// ---------------- graph preprocessing ----------------

__global__ void k_zero_i32(int* p, int n) {
    int i = blockIdx.x * blockDim.x + threadIdx.x;
    if (i < n) p[i] = 0;
}

__global__ void k_count(const long long* __restrict__ dst, int* __restrict__ cnt) {
    int e = blockIdx.x * blockDim.x + threadIdx.x;
    if (e < NEDGES) atomicAdd(&cnt[(int)dst[e]], 1);
}

// per-256-chunk exclusive scan; chunk totals to csum
__global__ void k_scanA(const int* __restrict__ cnt, int* __restrict__ off,
                        int* __restrict__ csum) {
    __shared__ int s[256];
    int i = blockIdx.x * 256 + threadIdx.x;
    int v = (i < NNODES) ? cnt[i] : 0;
    s[threadIdx.x] = v;
    __syncthreads();
    for (int d = 1; d < 256; d <<= 1) {
        int t = (threadIdx.x >= d) ? s[threadIdx.x - d] : 0;
        __syncthreads();
        s[threadIdx.x] += t;
        __syncthreads();
    }
    if (i < NNODES) off[i] = s[threadIdx.x] - v;   // exclusive within chunk
    if (threadIdx.x == 255) csum[blockIdx.x] = s[255];
}

// single-block exclusive scan of chunk totals
__global__ void k_scanB(const int* __restrict__ csum, int* __restrict__ coff) {
    __shared__ int s[256];
    int v = (threadIdx.x < NCHUNK) ? csum[threadIdx.x] : 0;
    s[threadIdx.x] = v;
    __syncthreads();
    for (int d = 1; d < 256; d <<= 1) {
        int t = (threadIdx.x >= d) ? s[threadIdx.x - d] : 0;
        __syncthreads();
        s[threadIdx.x] += t;
        __syncthreads();
    }
    coff[threadIdx.x] = s[threadIdx.x] - v;        // exclusive
}

// finalize offsets + inverse-sqrt degree (self-loop => deg >= 1)
__global__ void k_scanC(const int* __restrict__ cnt, int* __restrict__ off,
                        const int* __restrict__ coff, float* __restrict__ invs) {
    int i = blockIdx.x * 256 + threadIdx.x;
    if (i < NNODES) {
        off[i] += coff[i >> 8];
        invs[i] = rsqrtf((float)(cnt[i] + 1));
    }
    if (i == 0) off[NNODES] = NEDGES;
}

__global__ void k_fill(const long long* __restrict__ src, const long long* __restrict__ dst,
                       const int* __restrict__ off, int* __restrict__ cur,
                       int* __restrict__ csr) {
    int e = blockIdx.x * blockDim.x + threadIdx.x;
    if (e < NEDGES) {
        int d = (int)dst[e];
        int p = atomicAdd(&cur[d], 1);
        csr[off[d] + p] = (int)src[e];
    }
}

// ---------------- WMMA GEMM: C[M x NOUT] = A[M x K] @ B[K x NOUT] ----------------
// One wave per 16-row M tile, accumulating the full NP-wide strip (NP = NOUT
// padded to 16). 5 waves / block; B panel (16 x NP) staged via LDS in a
// K-pair-interleaved layout so each WMMA B fragment is ONE contiguous
// ds_load_b64 landing in an even-aligned VGPR pair (no mov shuffling).
// LDS slot for element (k, col) = ((k>>1)*NP + col)*2 + (k&1).
// Uses V_WMMA_F32_16X16X4_F32 (f32 in, f32 acc — full reference precision).
template <int K, int NP, int NOUT>
__global__ __launch_bounds__(160) void k_gemm(const float* __restrict__ A,
                                              const float* __restrict__ B,
                                              float* __restrict__ C) {
    __shared__ float lb[16 * NP];
    const int lane  = threadIdx.x & 31;
    const int wave  = threadIdx.x >> 5;
    const int mtile = blockIdx.x * 5 + wave;       // always < MTILES (exact fit)
    const int half  = lane >> 4;                   // 0: lanes 0-15, 1: lanes 16-31
    const int l16   = lane & 15;
    const int kbase = half * 2;                    // K offset 0 or 2 (A/B frag layout)
    const int row   = mtile * 16 + l16;

    constexpr int NT = NP / 16;
    v8f acc[NT] = {};

    for (int k0 = 0; k0 < K; k0 += 16) {
        __syncthreads();
        // stage B rows [k0, k0+16) into LDS, K-pair interleaved, pad cols with 0
        for (int idx = threadIdx.x; idx < 16 * NP; idx += 160) {
            int r = idx / NP, c = idx % NP;
            float v = (c < NOUT) ? B[(size_t)(k0 + r) * NOUT + c] : 0.0f;
            lb[(((r >> 1) * NP + c) << 1) | (r & 1)] = v;
        }
        __syncthreads();
#pragma unroll
        for (int kk = 0; kk < 16; kk += 4) {
            v2f a = *(const v2f*)(A + (size_t)row * K + k0 + kk + kbase);
            const int p = (kk + kbase) >> 1;       // K-pair index for this lane
#pragma unroll
            for (int t = 0; t < NT; t++) {
                v2f b = *(const v2f*)(lb + (((p * NP) + t * 16 + l16) << 1));
                acc[t] = __builtin_amdgcn_wmma_f32_16x16x4_f32(
                    false, a, false, b, (short)0, acc[t], false, false);
            }
        }
    }
    // C frag layout: VGPR v holds row M = half*8 + v, column = l16
#pragma unroll
    for (int t = 0; t < NT; t++) {
        int col = t * 16 + l16;
        if (col < NOUT) {
#pragma unroll
            for (int v = 0; v < 8; v++) {
                int m = mtile * 16 + half * 8 + v;
                C[(size_t)m * NOUT + col] = acc[t][v];
            }
        }
    }
}

// ---------------- aggregation (gather by dst) + bias (+ BN + ReLU) ----------------
// out[i] = f( invs[i] * ( invs[i]*H[i] + sum_{e: dst=i} invs[src]*H[src] ) + bias )
template <int F, int WPB, bool DO_BN>
__global__ void k_agg(const float* __restrict__ H, const int* __restrict__ off,
                      const int* __restrict__ csr, const float* __restrict__ invs,
                      const float* __restrict__ bias, const float* __restrict__ gamma,
                      const float* __restrict__ beta, const float* __restrict__ mean,
                      const float* __restrict__ var, float* __restrict__ out) {
    const int lane = threadIdx.x & 31;
    const int wave = threadIdx.x >> 5;
    const int node = blockIdx.x * WPB + wave;
    if (node >= NNODES) return;

    constexpr int VPL = (F + 127) / 128;   // float4 slots per lane
    float4 acc[VPL];
    const float wself = invs[node];
#pragma unroll
    for (int j = 0; j < VPL; j++) {
        int c = j * 128 + lane * 4;
        if (c < F) {
            float4 h = ld4(H + (size_t)node * F + c);
            acc[j] = make_float4(wself * h.x, wself * h.y, wself * h.z, wself * h.w);
        } else {
            acc[j] = make_float4(0.f, 0.f, 0.f, 0.f);
        }
    }
    const int s0 = off[node], s1 = off[node + 1];
    for (int e = s0; e < s1; e++) {
        int s  = csr[e];
        float w = invs[s];
#pragma unroll
        for (int j = 0; j < VPL; j++) {
            int c = j * 128 + lane * 4;
            if (c < F) {
                float4 h = ld4(H + (size_t)s * F + c);
                acc[j].x += w * h.x; acc[j].y += w * h.y;
                acc[j].z += w * h.z; acc[j].w += w * h.w;
            }
        }
    }
    const float scale = invs[node];
#pragma unroll
    for (int j = 0; j < VPL; j++) {
        int c = j * 128 + lane * 4;
        if (c < F) {
            float r[4] = {acc[j].x, acc[j].y, acc[j].z, acc[j].w};
            float4 o;
            float* po = &o.x;
#pragma unroll
            for (int q = 0; q < 4; q++) {
                int cc = c + q;
                float v = r[q] * scale + bias[cc];
                if (DO_BN) {
                    v = (v - mean[cc]) * rsqrtf(var[cc] + EPSBN) * gamma[cc] + beta[cc];
                    v = fmaxf(v, 0.0f);
                }
                po[q] = v;
            }
            *(float4*)(out + (size_t)node * F + c) = o;
        }
    }
}

// ---------------- host orchestration ----------------
extern "C" void kernel_launch(void* const* d_in, const int* in_sizes, int n_in,
                              void* d_out, int out_size, void* d_ws, size_t ws_size,
                              hipStream_t stream) {
    (void)in_sizes; (void)n_in; (void)out_size; (void)ws_size;
    const float*      x     = (const float*)d_in[0];
    const long long*  ei    = (const long long*)d_in[1];     // int64 per reference
    const long long*  esrc  = ei;                            // edge_index[0]
    const long long*  edst  = ei + NEDGES;                   // edge_index[1]
    const float* W1 = (const float*)d_in[2];
    const float* b1 = (const float*)d_in[3];
    const float* W2 = (const float*)d_in[4];
    const float* b2 = (const float*)d_in[5];
    const float* W3 = (const float*)d_in[6];
    const float* b3 = (const float*)d_in[7];
    const float* gm = (const float*)d_in[8];
    const float* bt = (const float*)d_in[9];
    const float* rm = (const float*)d_in[10];
    const float* rv = (const float*)d_in[11];
    float* outp = (float*)d_out;

    // carve workspace (~106 MB)
    char* w = (char*)d_ws;
    size_t o = 0;
    auto alloc = [&](size_t bytes) -> size_t {
        size_t r = o;
        o += (bytes + 255) & ~(size_t)255;
        return r;
    };
    int*   cnt  = (int*)(w + alloc((size_t)NNODES * 4));
    int*   cur  = (int*)(w + alloc((size_t)NNODES * 4));
    int*   off  = (int*)(w + alloc((size_t)(NNODES + 1) * 4));
    int*   csum = (int*)(w + alloc(256 * 4));
    int*   coff = (int*)(w + alloc(256 * 4));
    float* invs = (float*)(w + alloc((size_t)NNODES * 4));
    int*   csr  = (int*)(w + alloc((size_t)NEDGES * 4));
    float* Hbuf = (float*)(w + alloc((size_t)NNODES * HIDD * 4));
    float* Gbuf = (float*)(w + alloc((size_t)NNODES * HIDD * 4));

    const int EB = (NEDGES + 255) / 256;

    // ---- build CSR (by destination) + normalization ----
    k_zero_i32<<<(NNODES + 255) / 256, 256, 0, stream>>>(cnt, NNODES);
    k_zero_i32<<<(NNODES + 255) / 256, 256, 0, stream>>>(cur, NNODES);
    k_count<<<EB, 256, 0, stream>>>(edst, cnt);
    k_scanA<<<NCHUNK, 256, 0, stream>>>(cnt, off, csum);
    k_scanB<<<1, 256, 0, stream>>>(csum, coff);
    k_scanC<<<NCHUNK, 256, 0, stream>>>(cnt, off, coff, invs);
    k_fill<<<EB, 256, 0, stream>>>(esrc, edst, off, cur, csr);

    const int AGB = (NNODES + 7) / 8;

    // ---- layer 1: H = x @ W1 ; G = relu(BN(agg(H) + b1)) ----
    k_gemm<FEATD, HIDD, HIDD><<<MTILES / 5, 160, 0, stream>>>(x, W1, Hbuf);
    k_agg<HIDD, 8, true><<<AGB, 256, 0, stream>>>(Hbuf, off, csr, invs,
                                                  b1, gm, bt, rm, rv, Gbuf);
    // ---- layer 2: H = G @ W2 ; G = relu(BN(agg(H) + b2)) ----
    k_gemm<HIDD, HIDD, HIDD><<<MTILES / 5, 160, 0, stream>>>(Gbuf, W2, Hbuf);
    k_agg<HIDD, 8, true><<<AGB, 256, 0, stream>>>(Hbuf, off, csr, invs,
                                                  b2, gm, bt, rm, rv, Gbuf);
    // ---- layer 3: H3 = G @ W3 (pad 40->48 cols) ; out = agg(H3) + b3 ----
    k_gemm<HIDD, 48, CLSD><<<MTILES / 5, 160, 0, stream>>>(Gbuf, W3, Hbuf);
    k_agg<CLSD, 8, false><<<AGB, 256, 0, stream>>>(Hbuf, off, csr, invs,
                                                   b3, gm, bt, rm, rv, outp);
}